// YOLOv1Loss_20925080666761
// MI455X (gfx1250) — compile-verified
//
#include <hip/hip_runtime.h>
#include <math.h>

// ---------------- problem constants (from reference) ----------------
#define SS   7
#define HW   49          // S*S
#define BN   2           // boxes per cell
#define CC   20          // classes
#define MM   8           // max objects
#define CH   30          // B*5 + C channels
#define IMG  (CH * HW)   // 1470 floats per image

#define WAVES_PER_BLOCK 4
#define BLOCK_THREADS   (WAVES_PER_BLOCK * 32)

// ---------------- per-wave LDS layout (float offsets) ----------------
#define OFF_OUT   0
#define SZ_OUT    1472                         // 1470 padded to 16B multiple
#define OFF_IOUS  (OFF_OUT + SZ_OUT)           // [98][8]
#define OFF_MAX   (OFF_IOUS + HW * BN * MM)    // [98]
#define OFF_IT    (OFF_MAX + HW * BN)          // iou_target [98]
#define OFF_IM    (OFF_IT + HW * BN)           // iou_mask   [98]
#define OFF_BT    (OFF_IM + HW * BN)           // box_target [98][4]
#define OFF_BM    (OFF_BT + HW * BN * 4)       // box_mask   [98]
#define OFF_CT    (OFF_BM + HW * BN)           // class_target [49]
#define OFF_CM    (OFF_CT + HW)                // class_mask   [49]
#define OFF_TGT   (OFF_CM + HW)                // raw targets  [40]
#define WAVE_LDS  3200                         // padded stride (floats)

// ---------------- CDNA5 async global->LDS path ----------------
#if defined(__has_builtin)
#if __has_builtin(__builtin_amdgcn_global_load_async_to_lds_b64)
#define HAVE_ASYNC_LDS 1
#endif
#endif

#ifdef HAVE_ASYNC_LDS
typedef int v2i __attribute__((vector_size(2 * sizeof(int))));
typedef __attribute__((address_space(1))) void v_as1;
typedef __attribute__((address_space(3))) void v_as3;
typedef __attribute__((address_space(1))) v2i v2i_as1;
typedef __attribute__((address_space(3))) v2i v2i_as3;
#endif

__device__ __forceinline__ float sigm(float x) { return 1.0f / (1.0f + expf(-x)); }

__global__ __launch_bounds__(BLOCK_THREADS) void yolo_loss_main(
    const float* __restrict__ outputs, const float* __restrict__ targets,
    float* __restrict__ blockSums, int nImages)
{
    __shared__ float lds[WAVES_PER_BLOCK * WAVE_LDS];
    __shared__ float wsum[WAVES_PER_BLOCK];

    const int tid  = threadIdx.x;
    const int wave = tid >> 5;
    const int lane = tid & 31;
    const int img  = blockIdx.x * WAVES_PER_BLOCK + wave;
    float* W = lds + wave * WAVE_LDS;
    const bool active = (img < nImages);

    // ---------- Phase 0: stage output slab (1470 f32) + targets (40 f32) ----------
    if (active) {
        const float* gsrc = outputs + (size_t)img * IMG;
#ifdef HAVE_ASYNC_LDS
        for (int i = lane; i < IMG / 2; i += 32) {   // 735 x b64 chunks, 8B aligned
            __builtin_amdgcn_global_load_async_to_lds_b64(
                (v2i_as1*)(v_as1*)(void*)(gsrc + 2 * i),
                (v2i_as3*)(v_as3*)(void*)(W + OFF_OUT + 2 * i), 0, 0);
        }
#else
        for (int i = lane; i < IMG / 2; i += 32) {
            W[OFF_OUT + 2 * i]     = gsrc[2 * i];
            W[OFF_OUT + 2 * i + 1] = gsrc[2 * i + 1];
        }
#endif
        const float* tsrc = targets + (size_t)img * (MM * 5);
        for (int i = lane; i < MM * 5; i += 32) W[OFF_TGT + i] = tsrc[i];
#ifdef HAVE_ASYNC_LDS
#if __has_builtin(__builtin_amdgcn_s_wait_asynccnt)
        __builtin_amdgcn_s_wait_asynccnt(0);
#else
        asm volatile("s_wait_asynccnt 0" ::: "memory");
#endif
#endif
    }
    __syncthreads();

    // ---------- Phase 1: pred boxes, IoUs, max-IoU, mask init ----------
    int   numObj = 0;
    bool  anyPos = false;
    float gbx[MM], gby[MM], gbw[MM], gbh[MM], gcl[MM];
    bool  vld[MM];

    if (active) {
#pragma unroll
        for (int oi = 0; oi < MM; ++oi) {
            float t0 = W[OFF_TGT + oi * 5 + 0];
            float t1 = W[OFF_TGT + oi * 5 + 1];
            float t2 = W[OFF_TGT + oi * 5 + 2];
            float t3 = W[OFF_TGT + oi * 5 + 3];
            float t4 = W[OFF_TGT + oi * 5 + 4];
            vld[oi] = (t0 + t1 + t2 + t3 + t4) > 0.0f;
            numObj += vld[oi] ? 1 : 0;
            gbx[oi] = t0 * (float)SS;  gby[oi] = t1 * (float)SS;
            gbw[oi] = t2 * (float)SS;  gbh[oi] = t3 * (float)SS;
            gcl[oi] = t4;
        }
        for (int cell = lane; cell < HW; cell += 32) {
            int iy = cell / SS, ix = cell - iy * SS;
            float sg[8];
#pragma unroll
            for (int k = 0; k < 8; ++k) sg[k] = sigm(W[OFF_OUT + k * HW + cell]);
#pragma unroll
            for (int b = 0; b < BN; ++b) {
                float px = sg[b * 4 + 0] + (float)ix;
                float py = sg[b * 4 + 1] + (float)iy;
                float pw = sg[b * 4 + 2] * (float)SS;
                float ph = sg[b * 4 + 3] * (float)SS;
                float areaA = pw * ph;
                float ax0 = px - pw * 0.5f, ay0 = py - ph * 0.5f;
                float ax1 = px + pw * 0.5f, ay1 = py + ph * 0.5f;
                float mx = -3.0e38f;
#pragma unroll
                for (int oi = 0; oi < MM; ++oi) {
                    float bx0 = gbx[oi] - gbw[oi] * 0.5f, by0 = gby[oi] - gbh[oi] * 0.5f;
                    float bx1 = gbx[oi] + gbw[oi] * 0.5f, by1 = gby[oi] + gbh[oi] * 0.5f;
                    float tlx = fmaxf(ax0, bx0), tly = fmaxf(ay0, by0);
                    float brx = fminf(ax1, bx1), bry = fminf(ay1, by1);
                    float en  = (tlx < brx && tly < bry) ? 1.0f : 0.0f;
                    float ai  = (brx - tlx) * (bry - tly) * en;
                    float iou = ai / (areaA + gbw[oi] * gbh[oi] - ai);
                    float m   = vld[oi] ? iou : -1.0f;
                    W[OFF_IOUS + (cell * BN + b) * MM + oi] = m;
                    mx = fmaxf(mx, m);
                }
                W[OFF_MAX + cell * BN + b] = mx;
                anyPos = anyPos || (mx > 0.5f);
            }
        }
    }
    const int hasPos = __any(active && anyPos);   // wave32 ballot (per-image)

    if (active) {
        for (int cell = lane; cell < HW; cell += 32) {
#pragma unroll
            for (int b = 0; b < BN; ++b) {
                float mx = W[OFF_MAX + cell * BN + b];
                float m  = 1.0f;
                if (mx >= 0.5f && hasPos) m = 0.0f;
                if (numObj == 0)          m = 0.0f;
                W[OFF_IM + cell * BN + b] = m;
                W[OFF_IT + cell * BN + b] = 0.0f;
                W[OFF_BM + cell * BN + b] = 0.0f;
#pragma unroll
                for (int k = 0; k < 4; ++k) W[OFF_BT + (cell * BN + b) * 4 + k] = 0.0f;
            }
            W[OFF_CT + cell] = 0.0f;
            W[OFF_CM + cell] = 0.0f;
        }
    }
    __syncthreads();

    // ---------- Phase 2: sequential target build (lane 0, matches JAX loop order) ----------
    if (active && lane == 0) {
#pragma unroll 1
        for (int oi = 0; oi < MM; ++oi) {
            if (!vld[oi]) continue;
            float cx = floorf(gbx[oi]);
            float cy = floorf(gby[oi]);
            int cell = (int)(cy * (float)SS + cx);
            float ic0 = W[OFF_IOUS + (cell * BN + 0) * MM + oi];
            float ic1 = W[OFF_IOUS + (cell * BN + 1) * MM + oi];
            int amax = (ic1 > ic0) ? 1 : 0;            // argmax, tie -> 0
            float mi = W[OFF_MAX + cell * BN + amax];
            W[OFF_CT + cell] = gcl[oi];
            W[OFF_CM + cell] = 1.0f;
            int idx = cell * BN + amax;
            W[OFF_BT + idx * 4 + 0] = gbx[oi] - cx;
            W[OFF_BT + idx * 4 + 1] = gby[oi] - cy;
            W[OFF_BT + idx * 4 + 2] = gbw[oi] / (float)SS;
            W[OFF_BT + idx * 4 + 3] = gbh[oi] / (float)SS;
            W[OFF_BM + idx] = 1.0f;
            W[OFF_IT + idx] = mi;
            W[OFF_IM + idx] = 2.0f;
        }
    }
    __syncthreads();

    // ---------- Phase 3: loss terms ----------
    float lossPartial = 0.0f;
    if (active) {
        float objL = 0.f, noobjL = 0.f, xyL = 0.f, whL = 0.f, clsL = 0.f;
        for (int cell = lane; cell < HW; cell += 32) {
#pragma unroll
            for (int b = 0; b < BN; ++b) {
                float conf = sigm(W[OFF_OUT + (8 + b) * HW + cell]);
                int idx = cell * BN + b;
                float it = W[OFF_IT + idx];
                float im = W[OFF_IM + idx];
                float d  = conf - it;
                float sq = d * d;
                if (im == 2.0f)      objL   += sq;
                else if (im == 1.0f) noobjL += sq;
                if (W[OFF_BM + idx] > 0.0f) {
                    float sx = sigm(W[OFF_OUT + (b * 4 + 0) * HW + cell]);
                    float sy = sigm(W[OFF_OUT + (b * 4 + 1) * HW + cell]);
                    float sw = sigm(W[OFF_OUT + (b * 4 + 2) * HW + cell]);
                    float sh = sigm(W[OFF_OUT + (b * 4 + 3) * HW + cell]);
                    float b0 = W[OFF_BT + idx * 4 + 0], b1 = W[OFF_BT + idx * 4 + 1];
                    float b2 = W[OFF_BT + idx * 4 + 2], b3 = W[OFF_BT + idx * 4 + 3];
                    float dx = sx - b0, dy = sy - b1;
                    xyL += dx * dx + dy * dy;
                    float dw = sqrtf(sw) - sqrtf(b2);
                    float dh = sqrtf(sh) - sqrtf(b3);
                    whL += dw * dw + dh * dh;
                }
            }
            if (W[OFF_CM + cell] > 0.0f) {
                float mx = -3.0e38f;
#pragma unroll
                for (int c = 0; c < CC; ++c) mx = fmaxf(mx, W[OFF_OUT + (10 + c) * HW + cell]);
                float se = 0.f;
#pragma unroll
                for (int c = 0; c < CC; ++c) se += expf(W[OFF_OUT + (10 + c) * HW + cell] - mx);
                float lse = logf(se) + mx;
                int ct = (int)W[OFF_CT + cell];
                clsL += lse - W[OFF_OUT + (10 + ct) * HW + cell];
            }
        }
        lossPartial = 5.0f * (xyL + whL) + objL + 0.5f * noobjL + clsL;
    }

    // deterministic wave32 tree reduce
    for (int off = 16; off > 0; off >>= 1)
        lossPartial += __shfl_down(lossPartial, off, 32);
    if (lane == 0) wsum[wave] = lossPartial;
    __syncthreads();
    if (tid == 0) {
        float s = 0.f;
        for (int w = 0; w < WAVES_PER_BLOCK; ++w) s += wsum[w];
        blockSums[blockIdx.x] = s;
    }
}

// deterministic final reduction (fixed order, double accumulate)
__global__ __launch_bounds__(256) void yolo_loss_reduce(
    const float* __restrict__ blockSums, int n, float* __restrict__ out)
{
    __shared__ double sm[256];
    double s = 0.0;
    for (int i = threadIdx.x; i < n; i += 256) s += (double)blockSums[i];
    sm[threadIdx.x] = s;
    __syncthreads();
    for (int off = 128; off > 0; off >>= 1) {
        if ((int)threadIdx.x < off) sm[threadIdx.x] += sm[threadIdx.x + off];
        __syncthreads();
    }
    if (threadIdx.x == 0) out[0] = (float)sm[0];
}

extern "C" void kernel_launch(void* const* d_in, const int* in_sizes, int n_in,
                              void* d_out, int out_size, void* d_ws, size_t ws_size,
                              hipStream_t stream) {
    (void)n_in; (void)out_size; (void)ws_size;
    const float* outputs = (const float*)d_in[0];
    const float* targets = (const float*)d_in[1];
    float* out = (float*)d_out;
    int nImages = in_sizes[0] / IMG;
    int nBlocks = (nImages + WAVES_PER_BLOCK - 1) / WAVES_PER_BLOCK;
    float* blockSums = (float*)d_ws;

    yolo_loss_main<<<nBlocks, BLOCK_THREADS, 0, stream>>>(outputs, targets, blockSums, nImages);
    yolo_loss_reduce<<<1, 256, 0, stream>>>(blockSums, nBlocks, out);
}